// MultiHeadedEMA_17136919511675
// MI455X (gfx1250) — compile-verified
//
#include <hip/hip_runtime.h>

typedef float v2f __attribute__((ext_vector_type(2)));
typedef float v8f __attribute__((ext_vector_type(8)));

// ---- gfx1250 async-to-LDS path (probe via __has_builtin; fallback compiles) ----
#if defined(__has_builtin)
#  if __has_builtin(__builtin_amdgcn_global_load_async_to_lds_b128)
#    define EMA_ASYNC_LDS 1
#  endif
#  if __has_builtin(__builtin_amdgcn_s_wait_asynccnt)
#    define EMA_WAIT_BUILTIN 1
#  endif
#endif

#if defined(EMA_ASYNC_LDS)
#  if defined(EMA_WAIT_BUILTIN)
#    define WAIT_ASYNC(n) __builtin_amdgcn_s_wait_asynccnt(n)
#  else
#    define WAIT_ASYNC(n) asm volatile("s_wait_asynccnt " #n ::: "memory")
#  endif
#else
#  define WAIT_ASYNC(n) ((void)0)
#endif

// Pointer types matching the builtin's signature: 16B int vector pointee,
// global (AS1) source / LDS (AS3) destination.
typedef int v4i_g __attribute__((vector_size(16)));
typedef __attribute__((address_space(1))) v4i_g* gbl_v4i_p;
typedef __attribute__((address_space(3))) v4i_g* lds_v4i_p;

// Problem shape (fixed by setup_inputs): B=2, L=4096, D=512, H=8
constexpr int B_   = 2;
constexpr int L_   = 4096;
constexpr int D_   = 512;
constexpr int H_   = 8;
constexpr int T_   = 16;            // WMMA time tile (M)
constexpr int UF_  = 4;             // chunks per super-iteration
constexpr int ROWS_ = T_ * UF_;     // 64 time rows staged per iteration
constexpr int NIT_  = L_ / ROWS_;   // 64 super-iterations (sequential carry steps)
constexpr int DT_   = 16;           // d tile (N)

__device__ __forceinline__ float sigmoidf(float v) {
    return 1.0f / (1.0f + __expf(-v));
}

// Stage 16 bytes global -> LDS (async on gfx1250, sync fallback otherwise)
__device__ __forceinline__ void stage16(const float* gsrc, float* ldst) {
#if defined(EMA_ASYNC_LDS)
    __builtin_amdgcn_global_load_async_to_lds_b128(
        (gbl_v4i_p)(v4i_g*)gsrc,
        (lds_v4i_p)(v4i_g*)ldst,
        /*imm offset*/0, /*cpol*/0);
#else
    *(float4*)ldst = *(const float4*)gsrc;
#endif
}

__global__ __launch_bounds__(512)
void ema_wmma_kernel(const float* __restrict__ x,
                     const float* __restrict__ expansion,
                     const float* __restrict__ reduction,
                     const float* __restrict__ alphas,
                     const float* __restrict__ damps,
                     const float* __restrict__ ralphas,
                     const float* __restrict__ rdamps,
                     float* __restrict__ out)
{
    // [buf][dir][row][d] double-buffered stage: 16 KB; per-wave head tiles: 64 KB
    __shared__ __attribute__((aligned(16))) float stage[2][2][ROWS_][DT_];
    __shared__ __attribute__((aligned(16))) float red[16][ROWS_][DT_];

    const int blk = blockIdx.x;               // 0..63
    const int b   = blk / (D_ / DT_);
    const int d0  = (blk % (D_ / DT_)) * DT_;

    const int tid  = threadIdx.x;
    const int wave = tid >> 5;                // 0..15 (wave32)
    const int lane = tid & 31;
    const int half = lane >> 4;
    const int n    = lane & 15;               // d column / A row
    const int dirw = wave >> 3;               // 0 = fwd heads, 1 = rev heads
    const int h    = wave & 7;
    const int hIdx = dirw ? (H_ + h) : h;

    // Per-head EMA parameters (wave-uniform)
    const float a = sigmoidf(dirw ? ralphas[h] : alphas[h]);
    const float w = (1.0f - a) * sigmoidf(dirw ? rdamps[h] : damps[h]);
    // Fold expansion*reduction into one per-(h,d) coefficient (linearity)
    const float coef = expansion[hIdx * D_ + d0 + n] * reduction[hIdx * D_ + d0 + n];

    // Decay powers w^0..w^16
    float wp[17];
    wp[0] = 1.0f;
    #pragma unroll
    for (int i = 1; i <= 16; ++i) wp[i] = wp[i - 1] * w;

    // A-matrix: 16x16 lower-triangular decay, 4 K-slices of 4 (WMMA f32 16x16x4 layout)
    v2f A[4];
    {
        const int r = n;
        #pragma unroll
        for (int k = 0; k < 4; ++k) {
            const int s0 = 4 * k + 2 * half;
            A[k].x = (s0     <= r) ? a * wp[r - s0]     : 0.0f;
            A[k].y = (s0 + 1 <= r) ? a * wp[r - s0 - 1] : 0.0f;
        }
    }

    // Staging / writeout thread mapping: one 16B group per thread
    const int e    = tid & 255;
    const int srow = e >> 2;                  // 0..63
    const int sq   = (e & 3) * 4;             // d sub-offset {0,4,8,12}
    const int sdir = tid >> 8;                // threads 0-255 fwd, 256-511 rev

    const long xBase = (long)b * L_ * D_;

    auto stage_iter = [&](int it, int buf) {
        long l; int lr;
        if (sdir == 0) { l = (long)it * ROWS_ + srow;               lr = srow; }
        else           { l = (long)(L_ - (it + 1) * ROWS_) + srow;  lr = (ROWS_ - 1) - srow; } // time-reversed
        stage16(&x[xBase + l * D_ + d0 + sq], &stage[buf][sdir][lr][sq]);
    };

    stage_iter(0, 0);                         // prologue fill

    float carry = 0.0f;                       // per-column EMA state

    for (int it = 0; it < NIT_; ++it) {
        const int buf = it & 1;
        // Prefetch next super-tile into the other buffer, then wait for current one.
        if (it + 1 < NIT_) { stage_iter(it + 1, buf ^ 1); WAIT_ASYNC(1); }
        else               { WAIT_ASYNC(0); }
        __syncthreads();                      // stage(it) visible to all waves

        const float (*S)[DT_] = stage[buf][dirw];

        // 4 carry-independent 16x16 local scans (16 WMMAs), chained only by the rank-1 fix
        #pragma unroll
        for (int sub = 0; sub < UF_; ++sub) {
            const int rbase = sub * T_;
            v8f c = {};
            #pragma unroll
            for (int k = 0; k < 4; ++k) {
                v2f Bm;
                const int s0 = 4 * k + 2 * half;
                Bm.x = S[rbase + s0][n];
                Bm.y = S[rbase + s0 + 1][n];
                c = __builtin_amdgcn_wmma_f32_16x16x4_f32(
                        false, A[k], false, Bm, (short)0, c, false, false);
            }
            // y[t] += w^(t+1)*carry ; scale by coef ; stash per-head rows
            float p = wp[1 + 8 * half];
            float last = 0.0f;
            #pragma unroll
            for (int j = 0; j < 8; ++j) {     // C layout: VGPR j -> row m = j + 8*half
                const float y = c[j] + p * carry;
                p *= w;
                red[wave][rbase + j + 8 * half][n] = coef * y;
                if (j == 7) last = y;         // half==1,j==7 -> row 15 (incl. w^16*carry)
            }
            carry = __shfl(last, n + 16, 32); // broadcast row-15 state to both half-lanes
        }
        __syncthreads();                      // red complete

        // Deterministic head reduction + writeout: one float4 per thread
        {
            float4 s4 = make_float4(0.f, 0.f, 0.f, 0.f);
            const int wb = sdir * 8;
            #pragma unroll
            for (int ww = 0; ww < 8; ++ww) {
                const float4 v = *(const float4*)&red[wb + ww][srow][sq];
                s4.x += v.x; s4.y += v.y; s4.z += v.z; s4.w += v.w;
            }
            long l;
            if (sdir == 0) l = (long)it * ROWS_ + srow;
            else           l = (long)(L_ - 1) - (long)it * ROWS_ - srow;  // lmax - row
            float* op = &out[xBase + l * D_ + d0 + sq];
            // 64-row block p: fwd visits at it=p, rev at it=63-p (disjoint, never equal).
            // First visitor (it<32) overwrites poison; second accumulates.
            // The per-iteration barrier chain orders the two -> deterministic.
            if (it < NIT_ / 2) {
                *(float4*)op = s4;
            } else {
                atomicAdd(op + 0, s4.x); atomicAdd(op + 1, s4.y);
                atomicAdd(op + 2, s4.z); atomicAdd(op + 3, s4.w);
            }
        }
    }
}

extern "C" void kernel_launch(void* const* d_in, const int* in_sizes, int n_in,
                              void* d_out, int out_size, void* d_ws, size_t ws_size,
                              hipStream_t stream) {
    (void)in_sizes; (void)n_in; (void)d_ws; (void)ws_size; (void)out_size;
    const float* x         = (const float*)d_in[0];
    const float* expansion = (const float*)d_in[1];
    const float* reduction = (const float*)d_in[2];
    const float* alphas    = (const float*)d_in[3];
    const float* damps     = (const float*)d_in[4];
    const float* ralphas   = (const float*)d_in[5];
    const float* rdamps    = (const float*)d_in[6];
    float* out             = (float*)d_out;

    dim3 grid(B_ * (D_ / DT_));   // 64 blocks (one per batch x 16-wide d-tile)
    dim3 block(512);              // 16 waves: 8 fwd heads + 8 rev heads
    ema_wmma_kernel<<<grid, block, 0, stream>>>(x, expansion, reduction,
                                                alphas, damps, ralphas, rdamps, out);
}